// CRF_82489141887694
// MI455X (gfx1250) — compile-verified
//
#include <hip/hip_runtime.h>
#include <hip/hip_bf16.h>

// ---------------- problem constants (from reference) ----------------
constexpr int   K      = 11;        // real tags
constexpr int   S      = 13;        // + START/STOP
constexpr int   START  = 11;
constexpr int   STOPT  = 12;
constexpr int   Bsz    = 4096;
constexpr int   Tlen   = 1024;
constexpr float NEG    = -10000.0f;
constexpr float LOGMIN = -1.0e30f;  // clamp for log(0); keeps everything finite
constexpr float LOG2E  = 1.44269504088896340736f;
constexpr float LN2    = 0.69314718055994530942f;

// ---------------- tiling ----------------
constexpr int WAVES  = 2;           // waves per block, 1 tile of 16 batch rows per wave
constexpr int TC     = 16;          // time-chunk staged in LDS
constexpr int ROWLEN = TC * K;      // 176 floats per row per chunk (contiguous in memory)

typedef float v2f __attribute__((ext_vector_type(2)));
typedef float v8f __attribute__((ext_vector_type(8)));
typedef unsigned int u32x4 __attribute__((ext_vector_type(4)));
typedef int i32x4 __attribute__((ext_vector_type(4)));
typedef int i32x8 __attribute__((ext_vector_type(8)));

__device__ __forceinline__ float fast_exp(float x) {
    return __builtin_amdgcn_exp2f(x * LOG2E);
}
__device__ __forceinline__ float fast_log(float x) {   // clamped, never -inf/NaN for x>=0
    return fmaxf(__builtin_amdgcn_logf(x) * LN2, LOGMIN);
}

// Issue one TDM 2D tile load: 16 rows x ROWLEN floats, row stride Tlen*K elements.
__device__ __forceinline__ void tdm_load_chunk(const float* gsrc, unsigned lds_off) {
    unsigned long long ga = (unsigned long long)(uintptr_t)gsrc;
    u32x4 g0;
    g0[0] = 1u;                                   // count=1, user descriptor
    g0[1] = lds_off;                              // LDS byte address
    g0[2] = (unsigned)ga;                         // global_addr[31:0]
    g0[3] = (unsigned)(ga >> 32) | (2u << 30);    // global_addr[56:32] | type=2 (image)

    const unsigned td0 = (unsigned)(Tlen * K);    // tensor_dim0 (elements)
    const unsigned td1 = 16u;                     // tensor_dim1 (rows)
    i32x8 g1;
    g1[0] = (int)(2u << 16);                      // data_size=2 -> 4 bytes
    g1[1] = (int)((td0 & 0xffffu) << 16);         // tensor_dim0[15:0]  @ bits 63:48
    g1[2] = (int)((td0 >> 16) | ((td1 & 0xffffu) << 16));   // td0[31:16] | td1[15:0]
    g1[3] = (int)((td1 >> 16) | ((unsigned)ROWLEN << 16));  // td1[31:16] | tile_dim0
    g1[4] = (int)16u;                             // tile_dim1=16, tile_dim2=0
    g1[5] = (int)td0;                             // tensor_dim0_stride[31:0]
    g1[6] = 0;                                    // stride0[47:32] | stride1[15:0]
    g1[7] = 0;                                    // tensor_dim1_stride[47:16] (unused, 2D)

    i32x4 gz4 = {0, 0, 0, 0};                     // groups 2/3 unused (tile_dim2==0)
    i32x8 gz8 = {0, 0, 0, 0, 0, 0, 0, 0};         // trailing group (clang-23 6-arg form)
    __builtin_amdgcn_tensor_load_to_lds(g0, g1, gz4, gz4, gz8, 0);
}

// Precompute E = exp(Tmat) padded to 16x16 (exp(-10000) == 0 kills forbidden/pad lanes)
__global__ void crf_prep(const float* __restrict__ Tmat, float* __restrict__ E) {
    int i  = threadIdx.x;           // 256 threads
    int fr = i >> 4, to = i & 15;
    float v = 0.0f;
    if (fr < S && to < S) v = __expf(Tmat[fr * S + to]);
    E[i] = v;
}

// Forward algorithm: one wave32 per 16-row batch tile.
// a' = log(exp(a - rowmax) @ E) + rowmax + e_t   via V_WMMA_F32_16X16X4_F32.
// Emissions double-buffered in LDS via TDM (tensor_load_to_lds + s_wait_tensorcnt).
__global__ __launch_bounds__(32 * WAVES)
void crf_fwd(const float* __restrict__ e,
             const float* __restrict__ Tmat,
             const unsigned char* __restrict__ mask,
             const float* __restrict__ E,
             float* __restrict__ logZ) {
    __shared__ float eBuf[WAVES][2][16 * ROWLEN]; // ping-pong emission tiles
    __shared__ float ptBuf[WAVES][16 * 17];       // P^T, stride 17 = bank-conflict-free

    const int   wid  = threadIdx.x >> 5;
    const int   lane = threadIdx.x & 31;
    const int   hi   = lane >> 4;                 // wave half
    const int   ln   = lane & 15;
    const int   ne   = (ln < K) ? ln : (K - 1);   // clamped emission slot
    const float emw  = (ln < K) ? 1.0f : 0.0f;    // zero for pad/START/STOP columns
    const int   tile = blockIdx.x * WAVES + wid;
    const int   b0   = tile * 16;

    const unsigned ebase0 = (unsigned)(uintptr_t)(void*)&eBuf[wid][0][0];
    const unsigned ebase1 = (unsigned)(uintptr_t)(void*)&eBuf[wid][1][0];
    const float*   gbase  = e + (size_t)b0 * Tlen * K;

    // Constant B fragments of E: B[k][n], k = j + 2*hi, chunk kc covers k=4kc..4kc+3
    v2f bf[4];
#pragma unroll
    for (int kc = 0; kc < 4; ++kc) {
        bf[kc].x = E[(4 * kc + 0 + 2 * hi) * 16 + ln];
        bf[kc].y = E[(4 * kc + 1 + 2 * hi) * 16 + ln];
    }

    // State a in WMMA C/D layout: a[v] = a[row v+8*hi][state ln]
    float a[8];
#pragma unroll
    for (int v = 0; v < 8; ++v) a[v] = (ln == START) ? 0.0f : NEG;

    // kick off chunk 0's emission tile
    tdm_load_chunk(gbase, ebase0);

    constexpr int NCH = Tlen / TC;
    for (int ch = 0; ch < NCH; ++ch) {
        const int t0 = ch * TC;

        // prefetch next chunk via TDM into the other buffer, then wait for current
        if (ch + 1 < NCH) {
            tdm_load_chunk(gbase + (size_t)(t0 + TC) * K, (ch & 1) ? ebase0 : ebase1);
            __builtin_amdgcn_s_wait_tensorcnt((short)1);   // in-order: current tile done
        } else {
            __builtin_amdgcn_s_wait_tensorcnt((short)0);
        }
        const float* eb = &eBuf[wid][ch & 1][0];

        // ---- per-chunk mask bitmasks: lanes 0..15 build 16 bits for row=lane ----
        unsigned mybits = 0u;
        if (lane < 16) {
            const unsigned char* mrow = mask + (size_t)(b0 + lane) * Tlen + t0;
            uint4 mw = *(const uint4*)mrow;                // 16 bytes, 16B aligned
            unsigned w[4] = {mw.x, mw.y, mw.z, mw.w};
#pragma unroll
            for (int d = 0; d < 4; ++d)
#pragma unroll
                for (int j = 0; j < 4; ++j)
                    mybits |= (((w[d] >> (8 * j)) & 0xffu) ? 1u : 0u) << (4 * d + j);
        }
        unsigned mbits[8];
#pragma unroll
        for (int v = 0; v < 8; ++v)
            mbits[v] = __shfl((int)mybits, v + 8 * hi, 32);   // row v+8*hi's bits

        for (int tt = 0; tt < TC; ++tt) {
            // per-row max (row spans one 16-lane half) and P = exp(a - rowmax),
            // stored transposed: P[m][n] -> ptBuf[n*17 + m]
            float rm[8];
#pragma unroll
            for (int v = 0; v < 8; ++v) {
                float r = a[v];
#pragma unroll
                for (int off = 1; off < 16; off <<= 1)
                    r = fmaxf(r, __shfl_xor(r, off, 16));
                rm[v] = r;
                ptBuf[wid][ln * 17 + v + 8 * hi] = fast_exp(a[v] - r);
            }

            // C = P @ E : two independent 2-chains of wmma f32 16x16x4, then merge
            v8f c0 = {0.f, 0.f, 0.f, 0.f, 0.f, 0.f, 0.f, 0.f};
            v8f c1 = {0.f, 0.f, 0.f, 0.f, 0.f, 0.f, 0.f, 0.f};
            v2f af[4];
#pragma unroll
            for (int kc = 0; kc < 4; ++kc) {      // A[m=ln][k=4kc+j+2hi] = ptBuf[k*17+ln]
                af[kc].x = ptBuf[wid][(4 * kc + 0 + 2 * hi) * 17 + ln];
                af[kc].y = ptBuf[wid][(4 * kc + 1 + 2 * hi) * 17 + ln];
            }
            c0 = __builtin_amdgcn_wmma_f32_16x16x4_f32(false, af[0], false, bf[0], (short)0, c0, false, false);
            c1 = __builtin_amdgcn_wmma_f32_16x16x4_f32(false, af[1], false, bf[1], (short)0, c1, false, false);
            c0 = __builtin_amdgcn_wmma_f32_16x16x4_f32(false, af[2], false, bf[2], (short)0, c0, false, false);
            c1 = __builtin_amdgcn_wmma_f32_16x16x4_f32(false, af[3], false, bf[3], (short)0, c1, false, false);
            c0 = c0 + c1;

            // a' = log(C) + rowmax + emission, masked blend (fully branchless)
#pragma unroll
            for (int v = 0; v < 8; ++v) {
                int   m  = v + 8 * hi;
                float em = eb[m * ROWLEN + tt * K + ne] * emw;
                float sc = fast_log(c0[v]) + rm[v] + em;
                float mk = (float)((mbits[v] >> tt) & 1u);
                a[v] = fmaf(mk, sc - a[v], a[v]);
            }
        }
    }

    // logZ[b] = lse_k(a[b,k] + Tmat[k,STOP]) over real tags
#pragma unroll
    for (int v = 0; v < 8; ++v) ptBuf[wid][ln * 17 + v + 8 * hi] = a[v];
    if (lane < 16) {
        int   m  = lane;
        float mx = -3.0e38f;
        float vals[K];
#pragma unroll
        for (int k = 0; k < K; ++k) {
            vals[k] = ptBuf[wid][k * 17 + m] + Tmat[k * S + STOPT];
            mx = fmaxf(mx, vals[k]);
        }
        float s = 0.0f;
#pragma unroll
        for (int k = 0; k < K; ++k) s += fast_exp(vals[k] - mx);
        logZ[b0 + m] = fast_log(s) + mx;
    }
}

// Gold path score, one thread per batch row (gather-bound).
__global__ void crf_gold(const float* __restrict__ e,
                         const float* __restrict__ Tmat,
                         const int* __restrict__ tags,
                         const unsigned char* __restrict__ mask,
                         float* __restrict__ gold) {
    int b = blockIdx.x * blockDim.x + threadIdx.x;
    if (b >= Bsz) return;
    const int*           tg = tags + (size_t)b * Tlen;
    const unsigned char* mk = mask + (size_t)b * Tlen;
    const float*         eb = e + (size_t)b * Tlen * K;

    float em = 0.0f, tr = 0.0f;
    int   cnt = 0, prev = 0, g0 = 0;
    for (int t = 0; t < Tlen; ++t) {
        int g = tg[t];
        g = g < 0 ? 0 : (g > K - 1 ? K - 1 : g);
        float mf = mk[t] ? 1.0f : 0.0f;
        em += eb[t * K + g] * mf;
        if (t > 0) tr += Tmat[prev * S + g] * mf;   // maskf[:,1:]
        else g0 = g;
        cnt += mk[t] ? 1 : 0;
        prev = g;
    }
    int last = cnt - 1;
    if (last < 0) last = 0;
    int lt = tg[last];
    lt = lt < 0 ? 0 : (lt > K - 1 ? K - 1 : lt);
    gold[b] = Tmat[START * S + g0] + em + tr + Tmat[lt * S + STOPT];
}

// mean(logZ - gold)
__global__ void crf_reduce(const float* __restrict__ logZ,
                           const float* __restrict__ gold,
                           float* __restrict__ out) {
    __shared__ float sh[256];
    float s = 0.0f;
    for (int b = threadIdx.x; b < Bsz; b += 256) s += logZ[b] - gold[b];
    sh[threadIdx.x] = s;
    __syncthreads();
    for (int st = 128; st > 0; st >>= 1) {
        if (threadIdx.x < st) sh[threadIdx.x] += sh[threadIdx.x + st];
        __syncthreads();
    }
    if (threadIdx.x == 0) out[0] = sh[0] / (float)Bsz;
}

extern "C" void kernel_launch(void* const* d_in, const int* in_sizes, int n_in,
                              void* d_out, int out_size, void* d_ws, size_t ws_size,
                              hipStream_t stream) {
    const float*         e    = (const float*)d_in[0];
    const float*         Tmat = (const float*)d_in[1];
    const int*           tags = (const int*)d_in[2];
    const unsigned char* mask = (const unsigned char*)d_in[3];

    float* ws   = (float*)d_ws;
    float* E    = ws;                   // 256 floats
    float* logZ = ws + 256;             // 4096 floats
    float* gold = ws + 256 + Bsz;       // 4096 floats

    crf_prep<<<1, 256, 0, stream>>>(Tmat, E);
    crf_fwd<<<Bsz / 16 / WAVES, 32 * WAVES, 0, stream>>>(e, Tmat, mask, E, logZ);
    crf_gold<<<Bsz / 256, 256, 0, stream>>>(e, Tmat, tags, mask, gold);
    crf_reduce<<<1, 256, 0, stream>>>(logZ, gold, (float*)d_out);
}